// SSMLayer_83846351553269
// MI455X (gfx1250) — compile-verified
//
#include <hip/hip_runtime.h>
#include <hip/hip_bf16.h>

typedef __attribute__((ext_vector_type(2))) float v2f;
typedef __attribute__((ext_vector_type(8))) float v8f;

// ---------------- Model dims (compile-time) ----------------
#define BATCH     2
#define SEQLEN    2048
#define C_IN      256
#define D_MODEL   512
#define LOUT      1024           // after stride-2 conv
#define D_STATE   64
#define D_CONV    4
#define D_INNER   1024
#define HEADDIM   64
#define NHEADS    16
#define CONV_DIM  (D_INNER + 2*D_STATE)            // 1152
#define D_IN_PROJ (2*D_INNER + 2*D_STATE + NHEADS) // 2192
#define ROWS      (BATCH*LOUT)   // 2048
#define KSTEM     (C_IN * 7)     // 1792 (im2col K)

// ---------------- im2col for the stride-2 stem conv ----------------
// col[bl, c*7+k] = tokens[b, c, 2l-3+k]  (zero outside [0,SEQLEN))
__global__ void im2col_stem_kernel(const float* __restrict__ tokens,
                                   float* __restrict__ col) {
  const long idx = (long)blockIdx.x * 256 + threadIdx.x;
  if (idx >= (long)ROWS * KSTEM) return;
  const int j  = (int)(idx % KSTEM);
  const long bl = idx / KSTEM;
  const int c = j / 7;
  const int k = j - c * 7;
  const int l = (int)(bl & (LOUT - 1));
  const int b = (int)(bl >> 10);
  const int s = 2 * l - 3 + k;
  float v = 0.f;
  if (s >= 0 && s < SEQLEN) v = tokens[((long)b * C_IN + c) * SEQLEN + s];
  col[idx] = v;
}

// ---------------- fp32 WMMA GEMM: C[M,N] = A[M,K] * W[N,K]^T ----------------
// Each wave computes a 16x64 strip (4 C tiles): A operand loaded once per
// K-step and reused by 4 V_WMMA_F32_16X16X4_F32. Wave-uniform remainder path
// for N%64 != 0. Requires M%16==0, N%16==0, K%4==0.
__global__ void gemm_xt_wmma(const float* __restrict__ A,
                             const float* __restrict__ W,
                             float* __restrict__ C,
                             int M, int N, int K) {
  const int lane = threadIdx.x & 31;
  const int wave = threadIdx.x >> 5;
  const int tiles_n = N >> 4;
  const int strips  = (tiles_n + 3) >> 2;
  const int tile = blockIdx.x * (blockDim.x >> 5) + wave;
  if (tile >= (M >> 4) * strips) return;   // wave-uniform
  const int tm  = tile / strips;
  const int ts  = tile - tm * strips;
  const int tn0 = ts << 2;
  const int nt  = min(4, tiles_n - tn0);   // wave-uniform

  const int r    = lane & 15;
  const int koff = (lane >> 4) << 1;
  const float* arow = A + ((long)(tm * 16 + r)) * K + koff;
  const float* brow = W + ((long)(tn0 * 16 + r)) * K + koff;
  const long  bstep = (long)16 * K;        // next N tile of W

  if (nt == 4) {
    v8f acc0 = {0.f,0.f,0.f,0.f,0.f,0.f,0.f,0.f};
    v8f acc1 = acc0, acc2 = acc0, acc3 = acc0;
    #pragma unroll 2
    for (int k = 0; k < K; k += 4) {
      const v2f a  = *(const v2f*)(arow + k);
      const v2f b0 = *(const v2f*)(brow + k);
      const v2f b1 = *(const v2f*)(brow + bstep + k);
      const v2f b2 = *(const v2f*)(brow + 2 * bstep + k);
      const v2f b3 = *(const v2f*)(brow + 3 * bstep + k);
      acc0 = __builtin_amdgcn_wmma_f32_16x16x4_f32(false, a, false, b0, (short)0, acc0, false, false);
      acc1 = __builtin_amdgcn_wmma_f32_16x16x4_f32(false, a, false, b1, (short)0, acc1, false, false);
      acc2 = __builtin_amdgcn_wmma_f32_16x16x4_f32(false, a, false, b2, (short)0, acc2, false, false);
      acc3 = __builtin_amdgcn_wmma_f32_16x16x4_f32(false, a, false, b3, (short)0, acc3, false, false);
    }
    // C layout: VGPR i -> row i (lanes 0-15) / row i+8 (lanes 16-31), col = lane&15
    const int row0 = (lane >> 4) << 3;
    float* cbase = C + ((long)(tm * 16)) * N + tn0 * 16 + (lane & 15);
    #pragma unroll
    for (int i = 0; i < 8; ++i) {
      float* crow = cbase + (long)(row0 + i) * N;
      crow[0]  = acc0[i];
      crow[16] = acc1[i];
      crow[32] = acc2[i];
      crow[48] = acc3[i];
    }
  } else {
    for (int t = 0; t < nt; ++t) {         // wave-uniform trip count
      v8f acc = {0.f,0.f,0.f,0.f,0.f,0.f,0.f,0.f};
      const float* bt = brow + (long)t * bstep;
      #pragma unroll 4
      for (int k = 0; k < K; k += 4) {
        const v2f a = *(const v2f*)(arow + k);
        const v2f b = *(const v2f*)(bt + k);
        acc = __builtin_amdgcn_wmma_f32_16x16x4_f32(false, a, false, b, (short)0, acc, false, false);
      }
      const int row0 = (lane >> 4) << 3;
      float* cbase = C + ((long)(tm * 16)) * N + (tn0 + t) * 16 + (lane & 15);
      #pragma unroll
      for (int i = 0; i < 8; ++i)
        cbase[(long)(row0 + i) * N] = acc[i];
    }
  }
}

// ---------------- depthwise causal conv (width 4) + bias + SiLU ----------------
__global__ void dwconv_silu_kernel(const float* __restrict__ zx,
                                   const float* __restrict__ w,
                                   const float* __restrict__ bias,
                                   float* __restrict__ out) {
  const int c = blockIdx.x * 128 + threadIdx.x;
  if (c >= CONV_DIM) return;
  const long bl = blockIdx.y;           // 0..ROWS-1
  const int l = (int)(bl & (LOUT - 1)); // position within sequence
  const float* xcol = zx + bl * D_IN_PROJ + D_INNER + c;
  const float w0 = w[c * 4 + 0], w1 = w[c * 4 + 1], w2 = w[c * 4 + 2], w3 = w[c * 4 + 3];
  float acc = w3 * xcol[0];
  if (l >= 1) acc = fmaf(w2, xcol[-(long)D_IN_PROJ],     acc);
  if (l >= 2) acc = fmaf(w1, xcol[-(long)2 * D_IN_PROJ], acc);
  if (l >= 3) acc = fmaf(w0, xcol[-(long)3 * D_IN_PROJ], acc);
  const float v = acc + bias[c];
  out[bl * CONV_DIM + c] = v / (1.f + __expf(-v));
}

// ---------------- sequential SSM scan ----------------
// One workgroup per (b, head). 256 threads: thread t owns p=t>>2, n-chunk q=t&3
// State h[p, 16 n-values] in registers; per-step x/B/C staged in LDS.
__global__ void ssm_scan_kernel(const float* __restrict__ zx,
                                const float* __restrict__ xbc,
                                const float* __restrict__ dtb,
                                const float* __restrict__ alog,
                                const float* __restrict__ Dp,
                                float* __restrict__ y) {
  const int b  = blockIdx.x >> 4;
  const int hh = blockIdx.x & 15;
  const int tid = threadIdx.x;
  const int p = tid >> 2;
  const int q = tid & 3;
  __shared__ float sx[HEADDIM], sB[D_STATE], sC[D_STATE], sdt[2];
  const float Ah   = -__expf(alog[hh]);
  const float dtbh = dtb[hh];
  const float Dh   = Dp[hh];
  float hreg[16];
  #pragma unroll
  for (int i = 0; i < 16; ++i) hreg[i] = 0.f;

  for (int l = 0; l < LOUT; ++l) {
    const long bl = (long)b * LOUT + l;
    const float* row = xbc + bl * CONV_DIM;
    if (tid < 64) {
      sx[tid] = row[hh * HEADDIM + tid];
    } else if (tid < 128) {
      sB[tid - 64] = row[D_INNER + (tid - 64)];
    } else if (tid < 192) {
      sC[tid - 128] = row[D_INNER + D_STATE + (tid - 128)];
    } else if (tid == 192) {
      const float raw = zx[bl * D_IN_PROJ + (D_IN_PROJ - NHEADS) + hh] + dtbh;
      const float dtv = (raw > 20.f) ? raw : log1pf(__expf(raw)); // softplus
      sdt[0] = dtv;
      sdt[1] = __expf(dtv * Ah);                                  // dA
    }
    __syncthreads();
    const float dtv = sdt[0];
    const float dA  = sdt[1];
    const float xv  = sx[p];
    const float xdt = xv * dtv;
    const float* Bq = sB + q * 16;
    const float* Cq = sC + q * 16;
    float accv = 0.f;
    #pragma unroll
    for (int i = 0; i < 16; ++i) {
      hreg[i] = fmaf(hreg[i], dA, xdt * Bq[i]);
      accv    = fmaf(hreg[i], Cq[i], accv);
    }
    // reduce partial dot over the 4 n-chunks (lanes 4p..4p+3 within wave32)
    accv += __shfl_xor(accv, 1, 32);
    accv += __shfl_xor(accv, 2, 32);
    if (q == 0)
      y[bl * D_INNER + hh * HEADDIM + p] = fmaf(xv, Dh, accv);
    __syncthreads();
  }
}

// ---------------- gate with SiLU(z) + RMSNorm (in-place on y) ----------------
__global__ void gate_rmsnorm_kernel(const float* __restrict__ zx,
                                    const float* __restrict__ nw,
                                    float* __restrict__ y) {
  const long bl = blockIdx.x;
  const int tid = threadIdx.x;
  __shared__ float red[256];
  float yv[4];
  float ss = 0.f;
  #pragma unroll
  for (int j = 0; j < 4; ++j) {
    const int i = tid + j * 256;
    const float z = zx[bl * D_IN_PROJ + i];
    const float g = z / (1.f + __expf(-z));
    const float v = y[bl * D_INNER + i] * g;
    yv[j] = v;
    ss = fmaf(v, v, ss);
  }
  red[tid] = ss;
  __syncthreads();
  for (int st = 128; st > 0; st >>= 1) {
    if (tid < st) red[tid] += red[tid + st];
    __syncthreads();
  }
  const float scale = rsqrtf(red[0] * (1.f / (float)D_INNER) + 1e-5f);
  #pragma unroll
  for (int j = 0; j < 4; ++j) {
    const int i = tid + j * 256;
    y[bl * D_INNER + i] = yv[j] * scale * nw[i];
  }
}

// ---------------- host launcher ----------------
extern "C" void kernel_launch(void* const* d_in, const int* in_sizes, int n_in,
                              void* d_out, int out_size, void* d_ws, size_t ws_size,
                              hipStream_t stream) {
  const float* tokens     = (const float*)d_in[0];
  const float* conv_w     = (const float*)d_in[1];
  const float* in_proj_w  = (const float*)d_in[2];
  const float* conv1d_w   = (const float*)d_in[3];
  const float* conv1d_b   = (const float*)d_in[4];
  const float* dt_bias    = (const float*)d_in[5];
  const float* A_log      = (const float*)d_in[6];
  const float* D_param    = (const float*)d_in[7];
  const float* norm_w     = (const float*)d_in[8];
  const float* out_proj_w = (const float*)d_in[9];
  float* out = (float*)d_out;
  float* ws  = (float*)d_ws;

  float* u0   = ws;                                 // ROWS*D_MODEL   = 1,048,576
  float* u1   = u0 + (long)ROWS * D_MODEL;          // ROWS*D_MODEL
  float* zx   = u1 + (long)ROWS * D_MODEL;          // ROWS*D_IN_PROJ = 4,489,216
  float* xact = zx + (long)ROWS * D_IN_PROJ;        // ROWS*CONV_DIM  = 2,359,296
  float* ybuf = xact + (long)ROWS * CONV_DIM;       // ROWS*D_INNER   = 2,097,152
  float* col  = zx;  // im2col scratch (ROWS*KSTEM = 3,670,016) aliases zx:
                     // consumed by the stem GEMM strictly before zx is produced.
  (void)ws_size; (void)in_sizes; (void)n_in; (void)out_size;

  // stem: im2col then WMMA GEMM  u0(2048x512) = col(2048x1792) * conv_w(512x1792)^T
  {
    const long nel = (long)ROWS * KSTEM;
    im2col_stem_kernel<<<(int)((nel + 255) / 256), 256, 0, stream>>>(tokens, col);
    const int strips = ((D_MODEL / 16) + 3) / 4;   // 8
    const int tiles  = (ROWS / 16) * strips;
    gemm_xt_wmma<<<(tiles + 7) / 8, 256, 0, stream>>>(col, conv_w, u0, ROWS, D_MODEL, KSTEM);
  }

  const float* u_in = u0;
  for (int blk = 0; blk < 2; ++blk) {
    const float* ipw = in_proj_w  + (long)blk * D_IN_PROJ * D_MODEL;
    const float* cw  = conv1d_w   + (long)blk * CONV_DIM * D_CONV;
    const float* cb  = conv1d_b   + (long)blk * CONV_DIM;
    const float* dtb = dt_bias    + (long)blk * NHEADS;
    const float* al  = A_log      + (long)blk * NHEADS;
    const float* Dpp = D_param    + (long)blk * NHEADS;
    const float* nw  = norm_w     + (long)blk * D_INNER;
    const float* opw = out_proj_w + (long)blk * D_MODEL * D_INNER;

    // in_proj: zx = u_in(2048x512) * ipw(2192x512)^T
    {
      const int strips = ((D_IN_PROJ / 16) + 3) / 4; // 35 (34 full + 1 remainder)
      const int tiles  = (ROWS / 16) * strips;
      gemm_xt_wmma<<<(tiles + 7) / 8, 256, 0, stream>>>(u_in, ipw, zx, ROWS, D_IN_PROJ, D_MODEL);
    }
    // depthwise conv + SiLU
    dwconv_silu_kernel<<<dim3(CONV_DIM / 128, ROWS), 128, 0, stream>>>(zx, cw, cb, xact);
    // sequential scan: one workgroup per (batch, head)
    ssm_scan_kernel<<<BATCH * NHEADS, 256, 0, stream>>>(zx, xact, dtb, al, Dpp, ybuf);
    // gate + RMSNorm (in place)
    gate_rmsnorm_kernel<<<ROWS, 256, 0, stream>>>(zx, nw, ybuf);
    // out_proj: dst = ybuf(2048x1024) * opw(512x1024)^T
    float* dst = (blk == 0) ? u1 : out;
    {
      const int strips = ((D_MODEL / 16) + 3) / 4;   // 8
      const int tiles  = (ROWS / 16) * strips;
      gemm_xt_wmma<<<(tiles + 7) / 8, 256, 0, stream>>>(ybuf, opw, dst, ROWS, D_MODEL, D_INNER);
    }
    u_in = u1;
  }
}